// VSGNRPNHead_37649683316950
// MI455X (gfx1250) — compile-verified
//
#include <hip/hip_runtime.h>

#define BLK 256
#define IOU_THR 0.6f
#define DW_CLAMP 4.1351665567f /* log(1000/16) */

// One wave32 pulls 2*M floats (M==32 -> 256B) from global straight into LDS
// via the CDNA5 async global->LDS path, then waits on ASYNCcnt.
__device__ __forceinline__ void async_gt_to_lds(float* sh_gt, const float* gt_b,
                                                int tid, int M) {
    if (M == 32) {
        if (tid < 32) {
            unsigned lds = (unsigned)(uintptr_t)(void*)&sh_gt[0] + (unsigned)tid * 8u;
            unsigned go  = (unsigned)tid * 8u;
            unsigned long long base = (unsigned long long)(uintptr_t)gt_b;
            asm volatile(
                "global_load_async_to_lds_b64 %0, %1, %2\n\t"
                "s_wait_asynccnt 0"
                :
                : "v"(lds), "v"(go), "s"(base)
                : "memory");
        }
    } else {
        if (tid < 2 * M) sh_gt[tid] = gt_b[tid];
    }
}

// IoU exactly as reference. Only add/sub/min/max/div -> no FMA contraction,
// all ops correctly rounded, so values are bit-identical to jnp AND between
// the two passes. When inter==0 the reference yields +0.0 (uni>0 always since
// gt length >= 16), so the division is predicated away for ~97% of pairs.
__device__ __forceinline__ float iou1d(float amin, float amax, float lenA,
                                       float gmin, float gmax) {
    float inter = fminf(amax, gmax) - fmaxf(amin, gmin);
    inter = fmaxf(inter, 0.0f);
    float iou = 0.0f;
    if (inter > 0.0f) {
        float uni = fmaxf(lenA + (gmax - gmin) - inter, 0.0f);
        iou = inter / uni;
    }
    return iou;
}

__global__ void init_kernel(unsigned* __restrict__ best, double* __restrict__ accum,
                            int n) {
    int i = blockIdx.x * BLK + threadIdx.x;
    if (i < n) best[i] = 0u;
    if (i == 0) *accum = 0.0;
}

// Pass 1: best_per_gt[b][m] = max_n iou[n][m]  (float-as-uint atomic max; iou>=0).
// atomicMax with 0 against the 0-initialized cell is a no-op -> predicated.
__global__ __launch_bounds__(BLK) void colmax_kernel(
    const float* __restrict__ anchors, const float* __restrict__ gt_boxes,
    unsigned* __restrict__ best, int N, int M) {
    __shared__ float    sh_gt[64];
    __shared__ unsigned sh_best[32];
    const int b   = blockIdx.y;
    const int tid = threadIdx.x;
    async_gt_to_lds(sh_gt, gt_boxes + (size_t)b * M * 2, tid, M);
    if (tid < M) sh_best[tid] = 0u;
    __syncthreads();

    const int n = blockIdx.x * BLK + tid;
    if (n < N) {
        float2 a   = ((const float2*)anchors)[n];
        float lenA = a.y - a.x + 1.0f;
        for (int m = 0; m < M; ++m) {
            float2 g  = ((const float2*)sh_gt)[m];
            float iou = iou1d(a.x, a.y, lenA, g.x, g.y);
            if (iou > 0.0f) atomicMax(&sh_best[m], __float_as_uint(iou));
        }
    }
    __syncthreads();
    if (tid < M) atomicMax(&best[b * M + tid], sh_best[tid]);
}

// Pass 2: matcher + decode + GIoU + reduction. Decode/GIoU (and the preds
// load) are skipped when weight==0: the reference multiplies the (always
// finite) loss by 0 there, so this is numerically identical.
__global__ __launch_bounds__(BLK) void loss_kernel(
    const float* __restrict__ preds, const float* __restrict__ anchors,
    const float* __restrict__ gt_boxes, const int* __restrict__ gt_labels,
    const unsigned* __restrict__ best, double* __restrict__ accum,
    int N, int M) {
    __shared__ float sh_gt[64];
    __shared__ float sh_best[32];
    __shared__ int   sh_lab[32];
    __shared__ float sh_red[BLK];
    const int b   = blockIdx.y;
    const int tid = threadIdx.x;
    async_gt_to_lds(sh_gt, gt_boxes + (size_t)b * M * 2, tid, M);
    if (tid < M) {
        sh_best[tid] = __uint_as_float(best[b * M + tid]);
        sh_lab[tid]  = gt_labels[b * M + tid];
    }
    __syncthreads();

    const int n = blockIdx.x * BLK + tid;
    float contrib = 0.0f;
    if (n < N) {
        float2 a   = ((const float2*)anchors)[n];
        float lenA = a.y - a.x + 1.0f;
        float lmax = -__builtin_inff();
        int   larg = 0;
        bool  upd  = false;
        for (int m = 0; m < M; ++m) {
            float2 g  = ((const float2*)sh_gt)[m];
            float iou = iou1d(a.x, a.y, lenA, g.x, g.y);
            if (iou > lmax) { lmax = iou; larg = m; }   // first-index argmax
            upd = upd || (iou == sh_best[m]);            // low-quality rescue
        }
        int match = (lmax < IOU_THR) ? -1 : larg;
        if (upd) match = larg;
        bool pos = (match >= 0) && (sh_lab[match] > 0);

        if (__builtin_expect(pos, 0)) {
            int   mcl = match;
            float t1 = sh_gt[2 * mcl], t2 = sh_gt[2 * mcl + 1];

            // decode
            float2 p  = ((const float2*)preds)[(size_t)b * N + n];
            float exc = (a.y + a.x) * 0.5f;
            float dx  = p.x / 10.0f;
            float dw  = fminf(p.y / 5.0f, DW_CLAMP);
            float pc  = dx * lenA + exc;
            float pw  = expf(dw) * lenA;
            float pb0 = pc - 0.5f * (pw - 1.0f);
            float pb1 = pc + 0.5f * (pw - 1.0f);

            // GIoU loss
            float px1 = fminf(pb0, pb1), px2 = fmaxf(pb0, pb1);
            float pa  = px2 - px1;
            float ta  = t2 - t1;
            float x1i = fmaxf(px1, t1), x2i = fminf(px2, t2);
            float ai  = (x2i > x1i) ? (x2i - x1i) : 0.0f;
            float x1e = fminf(px1, t1), x2e = fmaxf(px2, t2);
            float ae  = x2e - x1e + 1e-7f;
            float au  = pa + ta - ai + 1e-7f;
            float ious  = ai / au;
            float gious = ious - (ae - au) / ae;
            contrib = 1.0f - gious;
        }
    }
    sh_red[tid] = contrib;
    __syncthreads();
    for (int s = BLK / 2; s > 0; s >>= 1) {
        if (tid < s) sh_red[tid] += sh_red[tid + s];
        __syncthreads();
    }
    if (tid == 0 && sh_red[0] != 0.0f) atomicAdd(accum, (double)sh_red[0]);
}

__global__ void final_kernel(const double* __restrict__ accum, float* __restrict__ out) {
    if (threadIdx.x == 0 && blockIdx.x == 0) out[0] = (float)(*accum);
}

extern "C" void kernel_launch(void* const* d_in, const int* in_sizes, int n_in,
                              void* d_out, int out_size, void* d_ws, size_t ws_size,
                              hipStream_t stream) {
    const float* preds     = (const float*)d_in[0];
    const float* anchors   = (const float*)d_in[1];
    const float* gt_boxes  = (const float*)d_in[2];
    const int*   gt_labels = (const int*)d_in[3];
    float* out = (float*)d_out;

    const int N  = in_sizes[1] / 2;          // 31744
    const int B  = in_sizes[0] / (2 * N);    // 64
    const int M  = in_sizes[3] / B;          // 32
    const int BM = B * M;

    // workspace: [BM u32 best_per_gt][8B-aligned double accumulator]
    unsigned* best  = (unsigned*)d_ws;
    size_t    off   = ((size_t)BM * sizeof(unsigned) + 7) & ~(size_t)7;
    double*   accum = (double*)((char*)d_ws + off);

    init_kernel<<<(BM + BLK - 1) / BLK, BLK, 0, stream>>>(best, accum, BM);

    dim3 grid((N + BLK - 1) / BLK, B);
    colmax_kernel<<<grid, BLK, 0, stream>>>(anchors, gt_boxes, best, N, M);
    loss_kernel<<<grid, BLK, 0, stream>>>(preds, anchors, gt_boxes, gt_labels,
                                          best, accum, N, M);
    final_kernel<<<1, 32, 0, stream>>>(accum, out);
}